// NodeFeatLayer_68453188763822
// MI455X (gfx1250) — compile-verified
//
#include <hip/hip_runtime.h>

// ---------------------------------------------------------------------------
// NodeFeatLayer fused pipeline for MI455X (gfx1250, wave32, WMMA f32 16x16x4)
//
//  Kernel 1 (node_film): h = relu((1+gamma)*LN(node@W_film+b) + beta), where
//                        [gamma|beta] = cond@W_cond + b_cond.   -> ws (N x 64)
//  Kernel 2 (edge_scatter): per edge e:
//                        ep  = tanh(edge_feats[mirror[e]] @ W_edge + b) * w[e]
//                        out[dst[e]] += h[src[e]] * ep        (atomicAdd f32)
//  edge GEMM fused into the scatter => edge_params never hits HBM (~410MB saved)
// ---------------------------------------------------------------------------

typedef __attribute__((ext_vector_type(2))) float v2f;
typedef __attribute__((ext_vector_type(8))) float v8f;

#define NODE_D 128
#define COND_D 256
#define EDGE_D 64
#define OUT_D  64
#define LN_EPS 1e-5f

static __device__ __forceinline__ v8f wmma_f32(v2f a, v2f b, v8f c) {
  // V_WMMA_F32_16X16X4_F32 : D(16x16,f32) = A(16x4,f32) x B(4x16,f32) + C
  return __builtin_amdgcn_wmma_f32_16x16x4_f32(
      /*neg_a=*/false, a, /*neg_b=*/false, b,
      /*c_mod=*/(short)0, c, /*reuse_a=*/false, /*reuse_b=*/false);
}

// ---------------------------------------------------------------------------
// Node kernel: 128 threads = 4 waves, each wave owns 16 nodes.
// ---------------------------------------------------------------------------
__global__ __launch_bounds__(128) void node_film_kernel(
    const float* __restrict__ node_feats, const float* __restrict__ cond,
    const float* __restrict__ W_cond, const float* __restrict__ b_cond,
    const float* __restrict__ W_film, const float* __restrict__ b_film,
    float* __restrict__ h_out, int N)
{
  __shared__ float sH[4 * 16 * 65];            // 16x64 tile per wave, padded

  const int tid  = threadIdx.x;
  const int wave = tid >> 5;
  const int lane = tid & 31;
  const int half = lane >> 4;                  // 0: K pair {k,k+1}, 1: {k+2,k+3}
  const int c    = lane & 15;                  // column (and A-row) id
  float* tile = &sH[wave * 16 * 65];

  const int nb    = blockIdx.x * 64 + wave * 16;
  const int garow = min(nb + c, N - 1);        // clamped A-row source

  // ---- Phase 1: hraw = node_feats @ W_film + b_film  -> LDS tile ----
  {
    const float* arow = node_feats + (size_t)garow * NODE_D;
    for (int t = 0; t < 4; ++t) {
      v8f acc = {0.f,0.f,0.f,0.f,0.f,0.f,0.f,0.f};
      const int col = t * 16 + c;
      for (int k = 0; k < NODE_D; k += 4) {
        v2f a = *(const v2f*)(arow + k + 2 * half);
        v2f b;
        b.x = W_film[(k + 2 * half)     * OUT_D + col];
        b.y = W_film[(k + 2 * half + 1) * OUT_D + col];
        acc = wmma_f32(a, b, acc);
      }
      const float bias = b_film[col];
#pragma unroll
      for (int i = 0; i < 8; ++i) {
        const int rowM = i + 8 * half;
        tile[rowM * 65 + col] = acc[i] + bias;
      }
    }
  }
  __syncthreads();

  // ---- Phase 2: LayerNorm per row (both half-waves handle row=c) ----
  {
    const int r = c;
    float mu = 0.f;
#pragma unroll
    for (int j = 0; j < OUT_D; ++j) mu += tile[r * 65 + j];
    mu *= (1.0f / OUT_D);
    float var = 0.f;
#pragma unroll
    for (int j = 0; j < OUT_D; ++j) {
      const float d = tile[r * 65 + j] - mu;
      var += d * d;
    }
    var *= (1.0f / OUT_D);
    const float rs = rsqrtf(var + LN_EPS);
#pragma unroll
    for (int j = 0; j < 32; ++j) {             // each half-wave writes 32 cols
      const int col = half * 32 + j;
      tile[r * 65 + col] = (tile[r * 65 + col] - mu) * rs;
    }
  }
  __syncthreads();

  // ---- Phase 3: gamma/beta = cond @ W_cond + b_cond; FiLM + relu ----
  {
    const float* crow = cond + (size_t)garow * COND_D;
    for (int t = 0; t < 4; ++t) {
      v8f ag = {0.f,0.f,0.f,0.f,0.f,0.f,0.f,0.f};
      v8f ab = {0.f,0.f,0.f,0.f,0.f,0.f,0.f,0.f};
      const int col = t * 16 + c;
      for (int k = 0; k < COND_D; k += 4) {
        v2f a = *(const v2f*)(crow + k + 2 * half);
        v2f bg, bb;
        bg.x = W_cond[(k + 2 * half)     * (2 * OUT_D) + col];
        bg.y = W_cond[(k + 2 * half + 1) * (2 * OUT_D) + col];
        bb.x = W_cond[(k + 2 * half)     * (2 * OUT_D) + col + OUT_D];
        bb.y = W_cond[(k + 2 * half + 1) * (2 * OUT_D) + col + OUT_D];
        ag = wmma_f32(a, bg, ag);
        ab = wmma_f32(a, bb, ab);
      }
      const float cg = b_cond[col] + 1.0f;     // gamma + 1
      const float cb = b_cond[col + OUT_D];
#pragma unroll
      for (int i = 0; i < 8; ++i) {
        const int rowM = i + 8 * half;
        const int g = nb + rowM;
        if (g < N) {
          const float hn  = tile[rowM * 65 + col];
          const float val = (ag[i] + cg) * hn + (ab[i] + cb);
          h_out[(size_t)g * OUT_D + col] = fmaxf(val, 0.f);
        }
      }
    }
  }
}

// ---------------------------------------------------------------------------
// Fused edge kernel: 256 threads = 8 waves, each wave owns 16 edges.
// GEMM A rows are gathered through `mirror`, so tanh(edge@W) never hits HBM.
// ---------------------------------------------------------------------------
__global__ __launch_bounds__(256) void edge_scatter_kernel(
    const float* __restrict__ edge_feats, const float* __restrict__ edge_weights,
    const int* __restrict__ src, const int* __restrict__ dst,
    const int* __restrict__ mirror,
    const float* __restrict__ W_edge, const float* __restrict__ b_edge,
    const float* __restrict__ h, float* __restrict__ out, int E)
{
  __shared__ float sW[EDGE_D * OUT_D];          // 16 KB staged weights

  const int tid = threadIdx.x;
  for (int i = tid; i < EDGE_D * OUT_D; i += 256) sW[i] = W_edge[i];
  __syncthreads();

  const int wave  = tid >> 5;
  const int lane  = tid & 31;
  const int half  = lane >> 4;
  const int c     = lane & 15;
  const int ebase = blockIdx.x * 128 + wave * 16;

  // --- preload full 16x64 A tile (rows = mirror[edge]) : 16 x float2/lane ---
  const int ea = min(ebase + c, E - 1);
  const int m  = mirror[ea];
  const float* arow = edge_feats + (size_t)m * EDGE_D;
  v2f a[16];
#pragma unroll
  for (int ks = 0; ks < 16; ++ks)
    a[ks] = *(const v2f*)(arow + ks * 4 + 2 * half);

  // --- per-output-row edge metadata (rows i+8*half of this wave's tile) ---
  int   srcv[8], dstv[8];
  float ewv[8];
#pragma unroll
  for (int i = 0; i < 8; ++i) {
    const int e = min(ebase + i + 8 * half, E - 1);
    srcv[i] = src[e];
    dstv[i] = dst[e];
    ewv[i]  = edge_weights[e];
  }

  for (int t = 0; t < 4; ++t) {
    v8f acc = {0.f,0.f,0.f,0.f,0.f,0.f,0.f,0.f};
    const int col = t * 16 + c;
#pragma unroll
    for (int ks = 0; ks < 16; ++ks) {
      v2f b;
      b.x = sW[(ks * 4 + 2 * half)     * OUT_D + col];
      b.y = sW[(ks * 4 + 2 * half + 1) * OUT_D + col];
      acc = wmma_f32(a[ks], b, acc);
    }
    const float bias = b_edge[col];
#pragma unroll
    for (int i = 0; i < 8; ++i) {
      const int e = ebase + i + 8 * half;
      if (e < E) {
        const float ep  = tanhf(acc[i] + bias) * ewv[i];
        const float msg = ep * h[(size_t)srcv[i] * OUT_D + col];
        atomicAdd(&out[(size_t)dstv[i] * OUT_D + col], msg);
      }
    }
  }
}

// ---------------------------------------------------------------------------
extern "C" void kernel_launch(void* const* d_in, const int* in_sizes, int n_in,
                              void* d_out, int out_size, void* d_ws, size_t ws_size,
                              hipStream_t stream)
{
  const float* node_feats = (const float*)d_in[0];
  const float* edge_feats = (const float*)d_in[1];
  const float* cond       = (const float*)d_in[2];
  const float* edge_w     = (const float*)d_in[3];
  const int*   src        = (const int*)d_in[4];
  const int*   dst        = (const int*)d_in[5];
  const int*   mirror     = (const int*)d_in[6];
  const float* W_edge     = (const float*)d_in[7];
  const float* b_edge     = (const float*)d_in[8];
  const float* W_cond     = (const float*)d_in[9];
  const float* b_cond     = (const float*)d_in[10];
  const float* W_film     = (const float*)d_in[11];
  const float* b_film     = (const float*)d_in[12];
  (void)n_in; (void)ws_size;

  const int N = in_sizes[0] / NODE_D;
  const int E = in_sizes[4];

  float* h_ws = (float*)d_ws;                  // N x 64 f32 (12.8 MB)
  float* out  = (float*)d_out;

  // out accumulates via atomics -> must start at zero every call
  hipMemsetAsync(out, 0, (size_t)out_size * sizeof(float), stream);

  node_film_kernel<<<dim3((N + 63) / 64), dim3(128), 0, stream>>>(
      node_feats, cond, W_cond, b_cond, W_film, b_film, h_ws, N);

  edge_scatter_kernel<<<dim3((E + 127) / 128), dim3(256), 0, stream>>>(
      edge_feats, edge_w, src, dst, mirror, W_edge, b_edge, h_ws, out, E);
}